// EnhancedGNN_59846074303179
// MI455X (gfx1250) — compile-verified
//
#include <hip/hip_runtime.h>
#include <cstdint>

// Problem constants (match reference)
#define NN      16384
#define IN_DIM  128
#define HID     256
#define EMB     128
#define NLAYER  3
#define NEDGE   262144
#define MASKW   (NN / 32)                 // 512 words per row
#define MASKB   ((size_t)NN * NN / 8)     // 32 MB
#define APAD    2                         // LDS row padding (floats) -> conflict-free
#define MAXK    256

typedef __attribute__((ext_vector_type(2))) float v2f;
typedef __attribute__((ext_vector_type(8))) float v8f;

// ---------------------------------------------------------------------------
// CDNA5 async global->LDS copy (ASYNCcnt-tracked), per CDNA5 ISA
// ---------------------------------------------------------------------------
__device__ __forceinline__ void async_load_b128(uint32_t lds_off, const float* gaddr) {
    asm volatile("global_load_async_to_lds_b128 %0, %1, off"
                 :: "v"(lds_off), "v"(gaddr) : "memory");
}
__device__ __forceinline__ void async_wait0() {
    asm volatile("s_wait_asynccnt 0x0" ::: "memory");
}
__device__ __forceinline__ uint32_t lds_addr32(const void* p) {
    // LDS aperture: low 32 bits of the flat pointer are the LDS byte offset
    return (uint32_t)(uintptr_t)p;
}

// Stage A tile [16 x K] (row mrow..mrow+15 of Ag) into LDS with row stride K+APAD
__device__ __forceinline__ void stage_A(float* smA, const float* Ag, int K,
                                        int mrow, int tid, int nthreads) {
    const int nvec = 16 * K / 4;                    // float4 count
    for (int i = tid; i < nvec; i += nthreads) {
        int m   = (i * 4) / K;
        int col = (i * 4) % K;
        async_load_b128(lds_addr32(smA + m * (K + APAD) + col),
                        Ag + (size_t)(mrow + m) * K + col);
    }
    async_wait0();
    __syncthreads();
}

// ---------------------------------------------------------------------------
// Adjacency bitmask build: bit(r,c)=1 for each edge, plus diagonal self-loops
// ---------------------------------------------------------------------------
__global__ __launch_bounds__(256) void build_adj(const int* __restrict__ ei,
                                                 uint32_t* __restrict__ mask) {
    int idx = blockIdx.x * blockDim.x + threadIdx.x;
    if (idx < NEDGE) {
        int r = ei[idx];
        int c = ei[NEDGE + idx];
        atomicOr(&mask[(size_t)r * MASKW + (c >> 5)], 1u << (c & 31));
    } else if (idx < NEDGE + NN) {
        int i = idx - NEDGE;
        atomicOr(&mask[(size_t)i * MASKW + (i >> 5)], 1u << (i & 31));
    }
}

// ---------------------------------------------------------------------------
// Row degrees -> d^-1/2  (reference: (rowsum + 1e-8)^-0.5)
// ---------------------------------------------------------------------------
__global__ __launch_bounds__(256) void degrees(const uint32_t* __restrict__ mask,
                                               float* __restrict__ dinv) {
    int i = blockIdx.x * blockDim.x + threadIdx.x;
    if (i >= NN) return;
    const uint32_t* row = mask + (size_t)i * MASKW;
    int cnt = 0;
    for (int w = 0; w < MASKW; ++w) cnt += __popc(row[w]);
    dinv[i] = rsqrtf((float)cnt + 1e-8f);
}

// ---------------------------------------------------------------------------
// SpMM: T[i,:] = dinv[i] * sum_j adj(i,j) * dinv[j] * H[j,:]
// One block (256 threads) per row; thread t owns column t.
// ---------------------------------------------------------------------------
__global__ __launch_bounds__(256) void spmm(const uint32_t* __restrict__ mask,
                                            const float* __restrict__ dinv,
                                            const float* __restrict__ Hin,
                                            float* __restrict__ Tout) {
    int i = blockIdx.x;
    int t = threadIdx.x;
    const uint32_t* row = mask + (size_t)i * MASKW;
    float acc = 0.f;
    for (int w = 0; w < MASKW; ++w) {
        uint32_t bits = row[w];           // uniform across the block -> scalarized
        while (bits) {
            int b = __ffs(bits) - 1;
            bits &= bits - 1;
            int j = (w << 5) + b;
            acc += dinv[j] * Hin[(size_t)j * HID + t];
        }
    }
    Tout[(size_t)i * HID + t] = dinv[i] * acc;
}

// ---------------------------------------------------------------------------
// WMMA f32 16x16 tile from LDS-staged A: C += A[16xK] * B[Kx16], K step 4.
// Lane mapping (wave32): l16 = lane&15, half = lane>>4.
//   A frag: a[v] = A[m=l16][k0 + 2*half + v]   (LDS, stride K+APAD)
//   B frag: b[v] = B[k0 + 2*half + v][n=l16]   (global, L2-resident weights)
//   C/D:    c[r] = C[r + 8*half][l16]
// ---------------------------------------------------------------------------
__device__ __forceinline__ v8f wmma_tile(const float* smA,
                                         const float* __restrict__ B,
                                         int K, int Ncols, int ncol, int lane) {
    const int half = lane >> 4;
    const int l16  = lane & 15;
    v8f c = {};
    const float* arow = smA + l16 * (K + APAD);
    for (int k0 = 0; k0 < K; k0 += 4) {
        int ka = k0 + 2 * half;
        v2f a, b;
        a.x = arow[ka];
        a.y = arow[ka + 1];
        b.x = B[(size_t)ka       * Ncols + ncol + l16];
        b.y = B[(size_t)(ka + 1) * Ncols + ncol + l16];
        c = __builtin_amdgcn_wmma_f32_16x16x4_f32(false, a, false, b,
                                                  (short)0, c, false, false);
    }
    return c;
}

// ---------------------------------------------------------------------------
// Generic GEMM + bias (+ optional leaky):  C[M,Ncols] = act(A[M,K]*B[K,Ncols]+b)
// block = (32,8): 8 waves, each one 16x16 tile. grid = (M/16, Ncols/128)
// ---------------------------------------------------------------------------
__global__ __launch_bounds__(256) void gemm_bias(const float* __restrict__ A,
                                                 const float* __restrict__ B,
                                                 const float* __restrict__ bias,
                                                 float* __restrict__ C,
                                                 int K, int Ncols, int leaky) {
    __shared__ float smA[16 * (MAXK + APAD)];
    const int lane = threadIdx.x;
    const int tid  = threadIdx.y * 32 + lane;
    const int mrow = blockIdx.x * 16;
    const int ncol = (blockIdx.y * 8 + threadIdx.y) * 16;

    __builtin_prefetch(B, 0, 2);                 // weights -> global_prefetch_b8
    stage_A(smA, A, K, mrow, tid, 256);          // async copy A tile to LDS

    v8f c = wmma_tile(smA, B, K, Ncols, ncol, lane);
    const int half = lane >> 4;
    const int l16  = lane & 15;
    const float bv = bias[ncol + l16];
#pragma unroll
    for (int r = 0; r < 8; ++r) {
        float v = c[r] + bv;
        if (leaky) v = v > 0.f ? v : 0.2f * v;
        C[(size_t)(mrow + r + 8 * half) * Ncols + ncol + l16] = v;
    }
}

// ---------------------------------------------------------------------------
// Fused conv layer: Hout = LayerNorm( leaky(T*W + b) + Hident )
// block = (32,8): 8 waves x 2 tiles = 16 n-tiles (HID=256). grid = N/16.
// ---------------------------------------------------------------------------
__global__ __launch_bounds__(256) void conv_layer(const float* __restrict__ T,
                                                  const float* __restrict__ W,
                                                  const float* __restrict__ bias,
                                                  const float* __restrict__ Hident,
                                                  const float* __restrict__ ln_g,
                                                  const float* __restrict__ ln_b,
                                                  float* __restrict__ Hout) {
    __shared__ float smA[16 * (HID + APAD)];
    __shared__ float sm[16][HID + 8];
    __shared__ float red1[16][16];
    __shared__ float red2[16][16];

    const int lane = threadIdx.x;
    const int wv   = threadIdx.y;
    const int tid  = wv * 32 + lane;
    const int half = lane >> 4;
    const int l16  = lane & 15;
    const int mrow = blockIdx.x * 16;

    __builtin_prefetch(W, 0, 2);
    stage_A(smA, T, HID, mrow, tid, 256);        // async copy T tile to LDS

#pragma unroll
    for (int t = 0; t < 2; ++t) {
        const int ncol = (wv * 2 + t) * 16;
        v8f c = wmma_tile(smA, W, HID, HID, ncol, lane);
        const float bv = bias[ncol + l16];
#pragma unroll
        for (int r = 0; r < 8; ++r) {
            int m = r + 8 * half;
            float v = c[r] + bv;
            v = v > 0.f ? v : 0.2f * v;                         // leaky
            v += Hident[(size_t)(mrow + m) * HID + ncol + l16]; // residual
            sm[m][ncol + l16] = v;
        }
    }
    __syncthreads();

    // LayerNorm: 256 threads -> (row, 16-col segment)
    const int row = tid >> 4;
    const int seg = tid & 15;
    float s = 0.f, s2 = 0.f;
#pragma unroll
    for (int c2 = 0; c2 < 16; ++c2) {
        float v = sm[row][seg * 16 + c2];
        s += v; s2 += v * v;
    }
    red1[row][seg] = s;
    red2[row][seg] = s2;
    __syncthreads();

    float mean = 0.f, msq = 0.f;
#pragma unroll
    for (int i = 0; i < 16; ++i) { mean += red1[row][i]; msq += red2[row][i]; }
    mean *= (1.f / HID);
    msq   = msq * (1.f / HID) - mean * mean;
    const float rstd = rsqrtf(msq + 1e-5f);

#pragma unroll
    for (int c2 = 0; c2 < 16; ++c2) {
        int col = seg * 16 + c2;
        float v = (sm[row][col] - mean) * rstd * ln_g[col] + ln_b[col];
        Hout[(size_t)(mrow + row) * HID + col] = v;
    }
}

// ---------------------------------------------------------------------------
extern "C" void kernel_launch(void* const* d_in, const int* in_sizes, int n_in,
                              void* d_out, int out_size, void* d_ws, size_t ws_size,
                              hipStream_t stream) {
    const float* x      = (const float*)d_in[0];
    const int*   ei     = (const int*)  d_in[1];
    const float* W_in   = (const float*)d_in[2];
    const float* b_in   = (const float*)d_in[3];
    const float* conv_W = (const float*)d_in[4];
    const float* conv_b = (const float*)d_in[5];
    const float* ln_g   = (const float*)d_in[6];
    const float* ln_b   = (const float*)d_in[7];
    const float* W_out  = (const float*)d_in[8];
    const float* b_out  = (const float*)d_in[9];
    float* out = (float*)d_out;

    char* ws = (char*)d_ws;
    uint32_t* mask = (uint32_t*)ws;                         // 32 MB
    size_t off = MASKB;
    float* dinv = (float*)(ws + off);  off += (size_t)NN * 4;
    off = (off + 255) & ~(size_t)255;
    float* H0   = (float*)(ws + off);  off += (size_t)NN * HID * 4;
    float* H1   = (float*)(ws + off);  off += (size_t)NN * HID * 4;
    float* T    = (float*)(ws + off);

    // 1) adjacency bitmask + degree normalization
    hipMemsetAsync(mask, 0, MASKB, stream);
    build_adj<<<(NEDGE + NN + 255) / 256, 256, 0, stream>>>(ei, mask);
    degrees<<<(NN + 255) / 256, 256, 0, stream>>>(mask, dinv);

    // 2) input projection: H0 = leaky(x @ W_in + b_in)
    gemm_bias<<<dim3(NN / 16, HID / 128), dim3(32, 8), 0, stream>>>(
        x, W_in, b_in, H0, IN_DIM, HID, 1);

    // 3) GCN layers
    float* hcur = H0;
    float* hnxt = H1;
    for (int l = 0; l < NLAYER; ++l) {
        spmm<<<NN, 256, 0, stream>>>(mask, dinv, hcur, T);
        conv_layer<<<NN / 16, dim3(32, 8), 0, stream>>>(
            T, conv_W + (size_t)l * HID * HID, conv_b + (size_t)l * HID,
            hcur, ln_g + (size_t)l * HID, ln_b + (size_t)l * HID, hnxt);
        float* tmp = hcur; hcur = hnxt; hnxt = tmp;
    }

    // 4) output projection: out = hcur @ W_out + b_out
    gemm_bias<<<dim3(NN / 16, EMB / 128), dim3(32, 8), 0, stream>>>(
        hcur, W_out, b_out, out, HID, EMB, 0);
}